// GraphConvGRU_82669530513658
// MI455X (gfx1250) — compile-verified
//
#include <hip/hip_runtime.h>
#include <math.h>

// Problem constants (from reference)
constexpr int B_   = 16;
constexpr int I_   = 128;
constexpr int H_   = 32;
constexpr int N_   = 2048;
constexpr int E_   = 32768;
constexpr int SEQ_ = 100;
constexpr int NT_  = 8;            // nodes per block (= waves per block)
constexpr int NH_  = N_ * H_;      // 65536
constexpr int PAD_ = 36;           // s_agg row stride: 16B-aligned rows, bank-conflict-free

typedef __attribute__((ext_vector_type(2))) float v2f;
typedef __attribute__((ext_vector_type(8))) float v8f;

// fast gate math: single-instruction transcendentals instead of precise-div / libm tanh
__device__ __forceinline__ float fast_sigmoid(float x) {
  return __builtin_amdgcn_rcpf(1.0f + __expf(-x));   // v_rcp_f32 + v_exp_f32
}
__device__ __forceinline__ float fast_tanh(float x) {
#if __has_builtin(__builtin_amdgcn_tanhf)
  return __builtin_amdgcn_tanhf(x);                  // v_tanh_f32 (gfx1250)
#else
  return 1.0f - 2.0f * __builtin_amdgcn_rcpf(__expf(2.0f * x) + 1.0f);
#endif
}

// ---------------------------------------------------------------- utilities
__global__ void k_zero_f32(float* __restrict__ p, int n) {
  int i = blockIdx.x * blockDim.x + threadIdx.x;
  if (i < n) p[i] = 0.0f;
}

__global__ void k_degrees(const int* __restrict__ src, const int* __restrict__ dst,
                          int* __restrict__ deg_out, int* __restrict__ deg_in) {
  int e = blockIdx.x * blockDim.x + threadIdx.x;
  if (e < E_) {
    atomicAdd(&deg_out[src[e]], 1);
    atomicAdd(&deg_in[dst[e]], 1);
  }
}

// serial exclusive scan over N_=2048 entries (one thread, trivial cost, deterministic)
__global__ void k_scan(const int* __restrict__ deg_in, int* __restrict__ row_ptr,
                       int* __restrict__ cursor) {
  if (blockIdx.x == 0 && threadIdx.x == 0) {
    int acc = 0;
    for (int n = 0; n < N_; ++n) {
      row_ptr[n] = acc;
      cursor[n]  = acc;
      acc += deg_in[n];
    }
    row_ptr[N_] = acc;
  }
}

__global__ void k_fill_csr(const int* __restrict__ src, const int* __restrict__ dst,
                           const int* __restrict__ deg_out, const int* __restrict__ deg_in,
                           int* __restrict__ cursor,
                           int* __restrict__ csr_src, float* __restrict__ csr_w) {
  int e = blockIdx.x * blockDim.x + threadIdx.x;
  if (e < E_) {
    int s = src[e], d = dst[e];
    float dout = (float)max(deg_out[s], 1);
    float din  = (float)max(deg_in[d], 1);
    float w    = rsqrtf(dout * din);
    int pos = atomicAdd(&cursor[d], 1);
    csr_src[pos] = s;
    csr_w[pos]   = w;
  }
}

// x-gates: out[b,hd] = dot(x[b,:I], W[:, hd]) + bias[hd]   (tiny: 512 dots of 128)
__global__ void k_xgate(const float* __restrict__ x, const float* __restrict__ W,
                        const float* __restrict__ bias, float* __restrict__ out) {
  int t = blockIdx.x * blockDim.x + threadIdx.x;
  if (t >= B_ * H_) return;
  int b = t >> 5, hd = t & 31;
  float acc = bias[hd];
  const float* xp = x + b * I_;
  const float* wp = W + hd;           // W is [I,H] row-major
  #pragma unroll 8
  for (int i = 0; i < I_; ++i) acc = fmaf(xp[i], wp[i * H_], acc);
  out[t] = acc;
}

// ---------------------------------------------------------------- fused GRU step
// grid = N_/NT_ blocks of 256 threads (8 wave32 waves); wave w owns node
// n = blockIdx.x*NT_ + w and the full 16(batch) x 32(H) tile for that node.
__global__ void __launch_bounds__(256)
k_step(const float* __restrict__ h_old, float* __restrict__ h_new,
       const int*  __restrict__ row_ptr, const int* __restrict__ csr_src,
       const float* __restrict__ csr_w,
       const float* __restrict__ Wg, const float* __restrict__ bg,
       const float* __restrict__ xr, const float* __restrict__ xz,
       const float* __restrict__ xh,
       float* __restrict__ out, int t) {
  __shared__ float s_agg[B_ * NT_][PAD_];     // rows r = w*16 + b; 144B rows: B128-aligned
  __shared__ float s_Wg[H_][H_ + 1];
  __shared__ float s_bg[H_];
  __shared__ float s_xr[B_][H_];
  __shared__ float s_xz[B_][H_];
  __shared__ float s_xh[B_][H_];

  const int tid  = threadIdx.x;
  const int wave = tid >> 5;
  const int lane = tid & 31;

  // stage constants into LDS
  for (int i = tid; i < H_ * H_; i += 256) s_Wg[i >> 5][i & 31] = Wg[i];
  if (tid < H_) s_bg[tid] = bg[tid];
  for (int i = tid; i < B_ * H_; i += 256) {
    s_xr[i >> 5][i & 31] = xr[i];
    s_xz[i >> 5][i & 31] = xz[i];
    s_xh[i >> 5][i & 31] = xh[i];
  }

  // ---- phase 1: CSR gather-aggregate, vectorized B128 per lane.
  // lane -> (batch quad q = lane>>3, 4-float H chunk off = (lane&7)*4)
  const int n     = blockIdx.x * NT_ + wave;
  const int start = row_ptr[n];
  const int end   = row_ptr[n + 1];
  const int q     = lane >> 3;
  const int off   = (lane & 7) * 4;

  float4 acc4[4];
  #pragma unroll
  for (int g = 0; g < 4; ++g) acc4[g] = make_float4(0.f, 0.f, 0.f, 0.f);

  for (int p = start; p < end; ++p) {
    const int   s = csr_src[p];
    const float w = csr_w[p];
    if (p + 1 < end)  // hint next source row toward the caches (global_prefetch_b8)
      __builtin_prefetch(h_old + (size_t)csr_src[p + 1] * H_ + off, 0, 0);
    const float* base = h_old + (size_t)s * H_ + off;
    #pragma unroll
    for (int g = 0; g < 4; ++g) {
      const int b = g * 4 + q;
      const float4 v = *reinterpret_cast<const float4*>(base + (size_t)b * NH_);
      acc4[g].x = fmaf(w, v.x, acc4[g].x);
      acc4[g].y = fmaf(w, v.y, acc4[g].y);
      acc4[g].z = fmaf(w, v.z, acc4[g].z);
      acc4[g].w = fmaf(w, v.w, acc4[g].w);
    }
  }
  #pragma unroll
  for (int g = 0; g < 4; ++g) {
    const int b = g * 4 + q;
    *reinterpret_cast<float4*>(&s_agg[wave * B_ + b][off]) = acc4[g];
  }

  __syncthreads();

  // ---- phase 2: h_conv = agg @ Wg via V_WMMA_F32_16X16X4_F32, then fused GRU.
  // A 16x4 f32 frag: lanes 0-15 -> M=lane, K = k0+{0,1}; lanes 16-31 -> K = k0+{2,3}.
  const int arow    = wave * B_ + (lane & 15);
  const int kbase   = (lane >> 4) * 2;
  const int halfsel = lane >> 4;

  #pragma unroll
  for (int c = 0; c < 2; ++c) {          // two 16-wide column tiles of H=32
    v8f C = {};
    const int bcol = c * 16 + (lane & 15);
    #pragma unroll
    for (int k0 = 0; k0 < H_; k0 += 4) {
      v2f a, bfr;
      a.x   = s_agg[arow][k0 + kbase];
      a.y   = s_agg[arow][k0 + kbase + 1];
      bfr.x = s_Wg[k0 + kbase][bcol];
      bfr.y = s_Wg[k0 + kbase + 1][bcol];
      C = __builtin_amdgcn_wmma_f32_16x16x4_f32(
              /*neg_a=*/false, a, /*neg_b=*/false, bfr,
              /*c_mod=*/(short)0, C, /*reuse_a=*/false, /*reuse_b=*/false);
    }
    // C frag layout: VGPR j -> row M = j + 8*halfsel, col = (lane&15) within tile.
    const int hd = bcol;                 // output H index
    #pragma unroll
    for (int j = 0; j < 8; ++j) {
      const int b  = j + 8 * halfsel;    // tile rows are exactly the batch index
      const float hc = C[j] + s_bg[hd];
      const float rg = fast_sigmoid(s_xr[b][hd] + hc);
      const float zg = fast_sigmoid(s_xz[b][hd] + hc);
      const float ht = fast_tanh(s_xh[b][hd] + rg * hc);
      const size_t idx = (size_t)b * NH_ + (size_t)n * H_ + hd;
      const float hn = (1.0f - zg) * h_old[idx] + zg * ht;
      h_new[idx] = hn;
      out[(size_t)b * ((size_t)SEQ_ * NH_) + (size_t)t * NH_ + (size_t)n * H_ + hd] = hn;
    }
  }
}

// ---------------------------------------------------------------- launcher
extern "C" void kernel_launch(void* const* d_in, const int* in_sizes, int n_in,
                              void* d_out, int out_size, void* d_ws, size_t ws_size,
                              hipStream_t stream) {
  (void)in_sizes; (void)n_in; (void)out_size; (void)ws_size;
  const float* x   = (const float*)d_in[0];
  const int*   src = (const int*)  d_in[1];
  const int*   dst = (const int*)  d_in[2];
  const float* Wr  = (const float*)d_in[3];
  const float* br  = (const float*)d_in[4];
  const float* Wz  = (const float*)d_in[5];
  const float* bz  = (const float*)d_in[6];
  const float* Wh  = (const float*)d_in[7];
  const float* bh  = (const float*)d_in[8];
  const float* Wg  = (const float*)d_in[9];
  const float* bg  = (const float*)d_in[10];
  float* out = (float*)d_out;

  // workspace carve-up (all 4-byte typed; base is 16B aligned)
  char* ws = (char*)d_ws;
  float* h0      = (float*)ws;                 ws += sizeof(float) * B_ * NH_;
  float* h1      = (float*)ws;                 ws += sizeof(float) * B_ * NH_;
  float* xr      = (float*)ws;                 ws += sizeof(float) * B_ * H_;
  float* xz      = (float*)ws;                 ws += sizeof(float) * B_ * H_;
  float* xh      = (float*)ws;                 ws += sizeof(float) * B_ * H_;
  int*   deg_out = (int*)ws;                   ws += sizeof(int) * N_;
  int*   deg_in  = (int*)ws;                   ws += sizeof(int) * N_;
  int*   row_ptr = (int*)ws;                   ws += sizeof(int) * (N_ + 1);
  int*   cursor  = (int*)ws;                   ws += sizeof(int) * N_;
  int*   csr_src = (int*)ws;                   ws += sizeof(int) * E_;
  float* csr_w   = (float*)ws;                 ws += sizeof(float) * E_;

  // preprocessing: degrees -> dst-CSR (removes f32 atomics from the hot loop)
  k_zero_f32<<<(B_ * NH_ + 255) / 256, 256, 0, stream>>>(h0, B_ * NH_);
  k_zero_f32<<<(2 * N_ + 255) / 256, 256, 0, stream>>>((float*)deg_out, 2 * N_);
  k_degrees<<<(E_ + 255) / 256, 256, 0, stream>>>(src, dst, deg_out, deg_in);
  k_scan<<<1, 1, 0, stream>>>(deg_in, row_ptr, cursor);
  k_fill_csr<<<(E_ + 255) / 256, 256, 0, stream>>>(src, dst, deg_out, deg_in,
                                                   cursor, csr_src, csr_w);
  // time-invariant input gates
  k_xgate<<<(B_ * H_ + 255) / 256, 256, 0, stream>>>(x, Wr, br, xr);
  k_xgate<<<(B_ * H_ + 255) / 256, 256, 0, stream>>>(x, Wz, bz, xz);
  k_xgate<<<(B_ * H_ + 255) / 256, 256, 0, stream>>>(x, Wh, bh, xh);

  // sequential scan: one fused kernel per step, ping-pong state buffers
  float* hcur = h0;
  float* hnxt = h1;
  for (int t = 0; t < SEQ_; ++t) {
    k_step<<<N_ / NT_, 256, 0, stream>>>(hcur, hnxt, row_ptr, csr_src, csr_w,
                                         Wg, bg, xr, xz, xh, out, t);
    float* tmp = hcur; hcur = hnxt; hnxt = tmp;
  }
}